// DLightGCN_78993038508476
// MI455X (gfx1250) — compile-verified
//
#include <hip/hip_runtime.h>

typedef float v2f __attribute__((ext_vector_type(2)));
typedef float v8f __attribute__((ext_vector_type(8)));

#define FDIM 128   // K * D = 4 * 32
#define F4DIM 32   // FDIM / 4

// ---------------- zero fill (float4 wide) ----------------
__global__ void k_zero_f4(float4* __restrict__ p, long n4) {
    long i = (long)blockIdx.x * blockDim.x + threadIdx.x;
    if (i < n4) p[i] = make_float4(0.f, 0.f, 0.f, 0.f);
}

// ---------------- x0 = concat(user_emb, item_emb) ----------------
__global__ void k_concat(const float4* __restrict__ ue, const float4* __restrict__ ie,
                         float4* __restrict__ x, long nu4, long tot4) {
    long i = (long)blockIdx.x * blockDim.x + threadIdx.x;
    if (i >= tot4) return;
    x[i] = (i < nu4) ? ue[i] : ie[i - nu4];
}

// ---------------- accumulate gathered batch rows: acc[b] += x[idx[b]] ----------------
__global__ void k_gather_accum(const float4* __restrict__ x,
                               const int* __restrict__ users, const int* __restrict__ items,
                               float4* __restrict__ uacc, float4* __restrict__ iacc,
                               int B, int nu) {
    int t = blockIdx.x * blockDim.x + threadIdx.x;
    int total = B * F4DIM;
    if (t >= 2 * total) return;
    const int* idxarr; float4* acc; int ofs;
    if (t < total) { idxarr = users; acc = uacc; ofs = 0; }
    else           { idxarr = items; acc = iacc; ofs = nu; t -= total; }
    int b = t >> 5, l = t & 31;
    long row = (long)idxarr[b] + ofs;
    float4 v = x[row * F4DIM + l];
    long o = (long)b * F4DIM + l;
    float4 a = acc[o];
    a.x += v.x; a.y += v.y; a.z += v.z; a.w += v.w;
    acc[o] = a;
}

// ---------------- SpMM scatter: y[dst] += val * x[src], one wave per edge ----------------
__global__ void k_spmm(const float4* __restrict__ x, const int* __restrict__ src,
                       const int* __restrict__ dst, const float* __restrict__ val,
                       float* __restrict__ y, int E) {
    int gid = blockIdx.x * blockDim.x + threadIdx.x;
    int e = gid >> 5;
    int lane = gid & 31;
    if (e >= E) return;
    int s = src[e];
    int d = dst[e];
    float v = val[e];
    // prefetch a future edge's source row (global_prefetch_b8) — hides gather latency
    int ep = e + 16;
    if (ep < E) __builtin_prefetch(&x[(long)src[ep] * F4DIM + lane], 0, 1);
    float4 m = x[(long)s * F4DIM + lane];
    float* yp = y + (long)d * FDIM + lane * 4;
    __hip_atomic_fetch_add(yp + 0, m.x * v, __ATOMIC_RELAXED, __HIP_MEMORY_SCOPE_AGENT);
    __hip_atomic_fetch_add(yp + 1, m.y * v, __ATOMIC_RELAXED, __HIP_MEMORY_SCOPE_AGENT);
    __hip_atomic_fetch_add(yp + 2, m.z * v, __ATOMIC_RELAXED, __HIP_MEMORY_SCOPE_AGENT);
    __hip_atomic_fetch_add(yp + 3, m.w * v, __ATOMIC_RELAXED, __HIP_MEMORY_SCOPE_AGENT);
}

// ---------------- batched dot via WMMA: out[b] = (1/16) * <Uacc[b], Iacc[b]> ----------------
// One wave per 16-row tile: the 16 dots are the diagonal of A(16x128) x B(128x16),
// computed as 32 chained v_wmma_f32_16x16x4_f32 (K chunks of 4), full fp32 precision.
// Rows are CLAMPED (not predicated) so all loads are unconditional b64 loads;
// two accumulators break the WMMA D->C dependency chain.
__global__ void k_dot_wmma(const float* __restrict__ U, const float* __restrict__ I,
                           float* __restrict__ out, int B) {
    int wave = (blockIdx.x * blockDim.x + threadIdx.x) >> 5;
    int lane = threadIdx.x & 31;
    int tiles = (B + 15) >> 4;
    if (wave >= tiles) return;
    int base = wave * 16;
    int m    = lane & 15;     // row within tile for both A and B fragments
    int half = lane >> 4;     // lanes 16-31 carry K=2,3 of each 4-chunk (ISA 7.12.2)
    long row = base + m;
    long rmax = (long)B - 1;
    if (row > rmax) row = rmax;            // clamp: duplicate loads, writes guarded below
    const v2f* up = (const v2f*)(U + row * FDIM);
    const v2f* ip = (const v2f*)(I + row * FDIM);

    v8f c0 = {}, c1 = {};
#pragma unroll
    for (int kc = 0; kc < F4DIM; kc += 2) {
        v2f a0 = up[kc * 2 + half];        // cols (4kc + 2*half, +1) of U row
        v2f b0 = ip[kc * 2 + half];        // B[k][n] = I[n][k] (transposed operand)
        v2f a1 = up[(kc + 1) * 2 + half];
        v2f b1 = ip[(kc + 1) * 2 + half];
        c0 = __builtin_amdgcn_wmma_f32_16x16x4_f32(
                false, a0, false, b0, (short)0, c0, false, false);
        c1 = __builtin_amdgcn_wmma_f32_16x16x4_f32(
                false, a1, false, b1, (short)0, c1, false, false);
    }
    v8f c = c0 + c1;

    // Diagonal (m,m): lanes 0-7 hold m=0..7 in c[m]; lanes 24-31 hold m=8..15 in c[lane-24].
    bool active = (lane < 8) || (lane >= 24);
    int vidx = (lane < 8) ? lane : (lane - 24);
    int mrow = (lane < 8) ? lane : (lane - 16);
    float dv = 0.f;
#pragma unroll
    for (int i = 0; i < 8; ++i) if (vidx == i) dv = c[i];
    if (active && (base + mrow) < B)
        out[base + mrow] = dv * 0.0625f;   // (1/4 layers-mean)^2 for user * item
}

extern "C" void kernel_launch(void* const* d_in, const int* in_sizes, int n_in,
                              void* d_out, int out_size, void* d_ws, size_t ws_size,
                              hipStream_t stream) {
    const float* ue   = (const float*)d_in[0];
    const float* ie   = (const float*)d_in[1];
    const int*   esrc = (const int*)d_in[2];
    const int*   edst = (const int*)d_in[3];
    const float* eval = (const float*)d_in[4];
    const int*   users = (const int*)d_in[5];
    const int*   items = (const int*)d_in[6];
    float* out = (float*)d_out;

    long nu = in_sizes[0] / FDIM;
    long ni = in_sizes[1] / FDIM;
    long n  = nu + ni;
    int  E  = in_sizes[2];
    int  B  = in_sizes[5];

    // workspace: two ping-pong node buffers + batch-row accumulators
    float* xa   = (float*)d_ws;
    float* xb   = xa + n * FDIM;
    float* uacc = xb + n * FDIM;
    float* iacc = uacc + (long)B * FDIM;

    const int TB = 256;
    long n4 = n * F4DIM;

    // x0 = concat(user_emb, item_emb)
    k_concat<<<(int)((n4 + TB - 1) / TB), TB, 0, stream>>>(
        (const float4*)ue, (const float4*)ie, (float4*)xa, nu * F4DIM, n4);

    // zero batch accumulators (uacc,iacc contiguous)
    long acc4 = (long)B * F4DIM * 2;
    k_zero_f4<<<(int)((acc4 + TB - 1) / TB), TB, 0, stream>>>((float4*)uacc, acc4);

    // layer-0 contribution
    long gthreads = (long)B * F4DIM * 2;
    int gblocks = (int)((gthreads + TB - 1) / TB);
    k_gather_accum<<<gblocks, TB, 0, stream>>>(
        (const float4*)xa, users, items, (float4*)uacc, (float4*)iacc, B, (int)nu);

    float* xc = xa;
    float* xn = xb;
    for (int l = 0; l < 3; ++l) {
        k_zero_f4<<<(int)((n4 + TB - 1) / TB), TB, 0, stream>>>((float4*)xn, n4);
        long sthreads = (long)E * 32;
        k_spmm<<<(int)((sthreads + TB - 1) / TB), TB, 0, stream>>>(
            (const float4*)xc, esrc, edst, eval, xn, E);
        k_gather_accum<<<gblocks, TB, 0, stream>>>(
            (const float4*)xn, users, items, (float4*)uacc, (float4*)iacc, B, (int)nu);
        float* t = xc; xc = xn; xn = t;
    }

    // batched diagonal dot on the matrix cores
    int tiles = (B + 15) / 16;
    long dthreads = (long)tiles * 32;
    k_dot_wmma<<<(int)((dthreads + TB - 1) / TB), TB, 0, stream>>>(uacc, iacc, out, B);
}